// Model_1245540515968
// MI455X (gfx1250) — compile-verified
//
#include <hip/hip_runtime.h>
#include <math.h>

#define N_ATOMS   10000
#define N_EDGES   200000
#define N_CH      16
#define N_RAD     8
#define K_MIX     128
#define N_STRUCT  128
#define A_PER_ATOM 960                     // 128 + 288 + 320 + 224
#define A_FLOATS  (N_ATOMS * A_PER_ATOM)   // 9,600,000 floats = 38.4 MB (L2-resident)
#define CUTOFF    5.0f
#define PI_F      3.14159265358979f

typedef __attribute__((ext_vector_type(2))) float v2f;
typedef __attribute__((ext_vector_type(8))) float v8f;

// ---------------------------------------------------------------------------
// Kernel 0: zero A-accumulator + per-atom dot accumulator + output
// ---------------------------------------------------------------------------
__global__ void zero_kernel(float* __restrict__ ws, float* __restrict__ out) {
    int idx = blockIdx.x * blockDim.x + threadIdx.x;
    const int total = A_FLOATS + N_ATOMS;
    if (idx < total) ws[idx] = 0.0f;
    if (idx < N_STRUCT) out[idx] = 0.0f;
}

// ---------------------------------------------------------------------------
// Kernel 1: edge kernel. One wave32 per edge. All lanes redundantly compute
// the cheap per-edge scalars (SIMD-free), then the 960 msg elements are
// scattered with global f32 atomics (A fits in the 192MB L2).
// A layout per atom: [l0: m0 k0..127][l1: m0..2 x k0..95][l2: 5x64][l3: 7x32]
// ---------------------------------------------------------------------------
__global__ void edge_kernel(const float* __restrict__ pos,
                            const float* __restrict__ emb,
                            const float* __restrict__ wr0,
                            const float* __restrict__ wr1,
                            const float* __restrict__ wr2,
                            const float* __restrict__ wr3,
                            const int*   __restrict__ species,
                            const int*   __restrict__ senders,
                            const int*   __restrict__ receivers,
                            float*       __restrict__ A) {
    const int e    = (blockIdx.x * blockDim.x + threadIdx.x) >> 5;
    const int lane = threadIdx.x & 31;
    if (e >= N_EDGES) return;

    const int snd = senders[e];
    const int rcv = receivers[e];

    const float dx = pos[rcv * 3 + 0] - pos[snd * 3 + 0];
    const float dy = pos[rcv * 3 + 1] - pos[snd * 3 + 1];
    const float dz = pos[rcv * 3 + 2] - pos[snd * 3 + 2];
    const float r     = sqrtf(dx * dx + dy * dy + dz * dz + 1e-12f);
    const float inv_r = 1.0f / r;
    const float x = dx * inv_r, y = dy * inv_r, z = dz * inv_r;

    // real spherical harmonics l = 0..3 (16 values)
    const float xy = x * y, yz = y * z, xz = x * z;
    const float x2 = x * x, y2 = y * y, z2 = z * z;
    float sh[16];
    sh[0]  = 0.28209479177387814f;
    sh[1]  = 0.4886025119029199f * y;
    sh[2]  = 0.4886025119029199f * z;
    sh[3]  = 0.4886025119029199f * x;
    sh[4]  = 1.0925484305920792f * xy;
    sh[5]  = 1.0925484305920792f * yz;
    sh[6]  = 0.31539156525252005f * (3.0f * z2 - 1.0f);
    sh[7]  = 1.0925484305920792f * xz;
    sh[8]  = 0.5462742152960396f * (x2 - y2);
    sh[9]  = 0.5900435899266435f * y * (3.0f * x2 - y2);
    sh[10] = 2.890611442640554f  * xy * z;
    sh[11] = 0.4570457994644658f * y * (5.0f * z2 - 1.0f);
    sh[12] = 0.3731763325901154f * z * (5.0f * z2 - 3.0f);
    sh[13] = 0.4570457994644658f * x * (5.0f * z2 - 1.0f);
    sh[14] = 1.445305721320277f  * z * (x2 - y2);
    sh[15] = 0.5900435899266435f * x * (x2 - 3.0f * y2);

    // radial basis: sin(n*pi*xr)/(xr+1e-3) * fcut via angle-addition recurrence
    const float xr = fminf(r * (1.0f / CUTOFF), 1.0f);
    float s1, c1;
    __sincosf(PI_F * xr, &s1, &c1);
    const float fcut  = 0.5f * (c1 + 1.0f);
    const float scale = fcut / (xr + 0.001f);
    float bess[N_RAD];
    float sprev = 0.0f, scur = s1;
    bess[0] = scur;
    #pragma unroll
    for (int n = 1; n < N_RAD; ++n) {
        float snext = 2.0f * c1 * scur - sprev;
        sprev = scur; scur = snext;
        bess[n] = scur;
    }
    #pragma unroll
    for (int n = 0; n < N_RAD; ++n) bess[n] *= scale;

    // R_l = bess @ w_rad_l  (w_rad_l is [N_RAD, N_MAX[l]] row-major)
    float R0[8], R1[6], R2[4], R3[2];
    #pragma unroll
    for (int k = 0; k < 8; ++k) {
        float a = 0.f;
        #pragma unroll
        for (int n = 0; n < 8; ++n) a += bess[n] * wr0[n * 8 + k];
        R0[k] = a;
    }
    #pragma unroll
    for (int k = 0; k < 6; ++k) {
        float a = 0.f;
        #pragma unroll
        for (int n = 0; n < 8; ++n) a += bess[n] * wr1[n * 6 + k];
        R1[k] = a;
    }
    #pragma unroll
    for (int k = 0; k < 4; ++k) {
        float a = 0.f;
        #pragma unroll
        for (int n = 0; n < 8; ++n) a += bess[n] * wr2[n * 4 + k];
        R2[k] = a;
    }
    #pragma unroll
    for (int k = 0; k < 2; ++k) {
        float a = 0.f;
        #pragma unroll
        for (int n = 0; n < 8; ++n) a += bess[n] * wr3[n * 2 + k];
        R3[k] = a;
    }

    const int spi = species[snd];
    float hj[N_CH];
    #pragma unroll
    for (int c = 0; c < N_CH; ++c) hj[c] = emb[spi * N_CH + c];

    float* Arow = A + (size_t)rcv * A_PER_ATOM;

    // l=0: 128 elems (m=0), kk = kr*16 + c
    for (int t = lane; t < 128; t += 32)
        atomicAdd(&Arow[t], R0[t >> 4] * hj[t & 15] * sh[0]);
    // l=1: 3*96 elems, base 128
    for (int t = lane; t < 288; t += 32) {
        int m = t / 96, kk = t - m * 96;
        atomicAdd(&Arow[128 + t], R1[kk >> 4] * hj[kk & 15] * sh[1 + m]);
    }
    // l=2: 5*64 elems, base 416
    for (int t = lane; t < 320; t += 32) {
        int m = t >> 6, kk = t & 63;
        atomicAdd(&Arow[416 + t], R2[kk >> 4] * hj[kk & 15] * sh[4 + m]);
    }
    // l=3: 7*32 elems, base 736
    for (int t = lane; t < 224; t += 32) {
        int m = t >> 5, kk = t & 31;
        atomicAdd(&Arow[736 + t], R3[kk >> 4] * hj[kk & 15] * sh[9 + m]);
    }
}

// ---------------------------------------------------------------------------
// Kernel 2: WMMA mixing einsum + fused epilogue.
// One wave = one 16-atom x 16-j tile. Full-precision V_WMMA_F32_16X16X4_F32.
// A-fragment (16x4 f32): lanes 0-15 -> K=k0,k0+1 ; lanes 16-31 -> K=k0+2,k0+3.
// B-fragment (4x16 f32): lane N = lane&15, rows mirrored with the half split.
// C/D: VGPR v holds M = v + 8*half, N = lane&15.
// ---------------------------------------------------------------------------
template <int KL, int BASE, int MCNT>
__device__ inline void mix_accum(const float* __restrict__ A,
                                 const float* __restrict__ W,
                                 int arow, int col, int half,
                                 float inv, float* bsum) {
    for (int m = 0; m < MCNT; ++m) {
        const float* Ar = A + (size_t)arow * A_PER_ATOM + BASE + m * KL + 2 * half;
        v8f c = {0.f, 0.f, 0.f, 0.f, 0.f, 0.f, 0.f, 0.f};
        #pragma unroll 8
        for (int k0 = 0; k0 < KL; k0 += 4) {
            v2f a;
            a.x = Ar[k0];
            a.y = Ar[k0 + 1];
            const int kb = k0 + 2 * half;
            v2f b;
            b.x = W[kb * K_MIX + col];
            b.y = W[(kb + 1) * K_MIX + col];
            c = __builtin_amdgcn_wmma_f32_16x16x4_f32(
                    false, a, false, b, (short)0, c, false, false);
        }
        #pragma unroll
        for (int v = 0; v < 8; ++v) { float e = c[v]; bsum[v] += e * e * inv; }
    }
}

__global__ void mix_kernel(const float* __restrict__ A,
                           const float* __restrict__ wm0,
                           const float* __restrict__ wm1,
                           const float* __restrict__ wm2,
                           const float* __restrict__ wm3,
                           const float* __restrict__ emb2,
                           const float* __restrict__ w_out,
                           const int*   __restrict__ species,
                           float*       __restrict__ atom_dot) {
    const int lane = threadIdx.x & 31;
    const int M    = lane & 15;
    const int half = lane >> 4;                     // 0 or 1
    const int jbase = (threadIdx.x >> 5) * 16;      // wave id -> j tile (8 tiles)
    const int col   = jbase + M;
    const int arow  = blockIdx.x * 16 + M;          // A-matrix row for this lane

    float bsum[8] = {0.f, 0.f, 0.f, 0.f, 0.f, 0.f, 0.f, 0.f};

    mix_accum<128,   0, 1>(A, wm0, arow, col, half, 1.0f,           bsum);
    mix_accum< 96, 128, 3>(A, wm1, arow, col, half, 0.57735026919f, bsum);
    mix_accum< 64, 416, 5>(A, wm2, arow, col, half, 0.44721359549f, bsum);
    mix_accum< 32, 736, 7>(A, wm3, arow, col, half, 0.37796447301f, bsum);

    // bsum[v] = B[blockIdx.x*16 + v + 8*half][col]  (complete over all l,m)
    const float wj = w_out[col];
    #pragma unroll
    for (int v = 0; v < 8; ++v) {
        const int atom = blockIdx.x * 16 + v + 8 * half;
        const float Bv = bsum[v];
        float h = Bv * Bv * emb2[species[atom] * K_MIX + col] * wj;
        // reduce over the 16 lanes of this half-wave (same atom row, j 0..15)
        #pragma unroll
        for (int off = 1; off < 16; off <<= 1) h += __shfl_xor(h, off, 32);
        if (M == 0) atomicAdd(&atom_dot[atom], h);
    }
}

// ---------------------------------------------------------------------------
// Kernel 3: per-atom scale + species bias, then batch segment-sum
// ---------------------------------------------------------------------------
__global__ void final_kernel(const float* __restrict__ atom_dot,
                             const float* __restrict__ scaling,
                             const float* __restrict__ comp_w,
                             const int*   __restrict__ species,
                             const int*   __restrict__ batch_seg,
                             float*       __restrict__ out) {
    const int n = blockIdx.x * blockDim.x + threadIdx.x;
    if (n >= N_ATOMS) return;
    const float v = atom_dot[n] * scaling[0] + comp_w[species[n]];
    atomicAdd(&out[batch_seg[n]], v);
}

// ---------------------------------------------------------------------------
extern "C" void kernel_launch(void* const* d_in, const int* in_sizes, int n_in,
                              void* d_out, int out_size, void* d_ws, size_t ws_size,
                              hipStream_t stream) {
    (void)in_sizes; (void)n_in; (void)out_size; (void)ws_size;
    // setup_inputs() dict order (w_rad / w_mix interleaved per l)
    const float* pos   = (const float*)d_in[0];
    const float* emb   = (const float*)d_in[1];
    const float* wr0   = (const float*)d_in[2];
    const float* wm0   = (const float*)d_in[3];
    const float* wr1   = (const float*)d_in[4];
    const float* wm1   = (const float*)d_in[5];
    const float* wr2   = (const float*)d_in[6];
    const float* wm2   = (const float*)d_in[7];
    const float* wr3   = (const float*)d_in[8];
    const float* wm3   = (const float*)d_in[9];
    const float* emb2  = (const float*)d_in[10];
    const float* wout  = (const float*)d_in[11];
    const float* compw = (const float*)d_in[12];
    const float* scal  = (const float*)d_in[13];
    const int* species = (const int*)d_in[14];
    const int* senders = (const int*)d_in[15];
    const int* recvs   = (const int*)d_in[16];
    const int* bseg    = (const int*)d_in[17];
    float* out = (float*)d_out;

    float* A        = (float*)d_ws;          // 38.4 MB
    float* atom_dot = A + A_FLOATS;          // 40 KB

    const int ztotal = A_FLOATS + N_ATOMS;
    zero_kernel<<<(ztotal + 255) / 256, 256, 0, stream>>>(A, out);

    // one wave32 per edge: 200000 * 32 threads / 256 = 25000 blocks
    edge_kernel<<<(N_EDGES * 32) / 256, 256, 0, stream>>>(
        pos, emb, wr0, wr1, wr2, wr3, species, senders, recvs, A);

    // 625 atom-tiles x 8 waves (j-tiles); exactly covers 10000 atoms, no guard
    mix_kernel<<<N_ATOMS / 16, 256, 0, stream>>>(
        A, wm0, wm1, wm2, wm3, emb2, wout, species, atom_dot);

    final_kernel<<<(N_ATOMS + 255) / 256, 256, 0, stream>>>(
        atom_dot, scal, compw, species, bseg, out);
}